// LocallyConnected2d_15779709846211
// MI455X (gfx1250) — compile-verified
//
#include <hip/hip_runtime.h>
#include <hip/hip_bf16.h>

typedef __attribute__((ext_vector_type(2))) float v2f;
typedef __attribute__((ext_vector_type(8))) float v8f;

#define TB        512   // 16 waves of 32
#define KC        72    // K-chunk: 8 channels x 9 taps
#define NCHUNK    8     // 576 / 72
#define RS        74    // LDS row stride (72 data + 2 pad floats)
#define ELEMS     9     // per-thread staged elements per chunk: 64*72/512

// out[b,oc,h,w] = sum_{c,i,j} x[b,c,w+i-1,h+j-1] * W[h,w,oc,c,i,j]
__global__ __launch_bounds__(TB) void lc2d_wmma_f32(
    const float* __restrict__ x,    // (64,64,32,32)
    const float* __restrict__ wts,  // (32,32,64,64,3,3)
    float* __restrict__ out)        // (64,64,32,32)
{
    __shared__ __align__(16) float As[64 * RS];  // A: [batch][k-in-chunk]
    __shared__ __align__(16) float Bs[64 * RS];  // W: [oc][k-in-chunk]

    const int tid = (int)threadIdx.x;
    const int bid = (int)blockIdx.x;   // spatial site = h*32 + w
    const int h   = bid >> 5;
    const int w   = bid & 31;

    const float* wbase = wts + (size_t)bid * 36864u;  // W[h][w] is contiguous

    // ---- chunk-invariant gather precompute -------------------------------
    int      xoff[ELEMS];   // clamped (always in-bounds) offset into x
    int      goff[ELEMS];   // offset into wbase (k0 advances by KC per chunk)
    int      loff[ELEMS];   // shared LDS layout offset for As/Bs
    unsigned vmask = 0;     // padding-validity bits
#pragma unroll
    for (int s = 0; s < ELEMS; ++s) {
        int e   = tid + s * TB;        // 0..4607 over the 64x72 chunk tile
        int row = e / KC;              // batch (A-side) == oc (B-side)
        int kk  = e - row * KC;        // 0..71 within chunk
        int q   = kk / 9;              // channel within 8-channel chunk
        int r   = kk - q * 9;
        int i   = r / 3;
        int j   = r - i * 3;
        int u   = w + i - 1;           // x dim-2 index (transposed patch!)
        int v   = h + j - 1;           // x dim-3 index
        bool ok = ((unsigned)u < 32u) && ((unsigned)v < 32u);
        int uc  = u < 0 ? 0 : (u > 31 ? 31 : u);
        int vc  = v < 0 ? 0 : (v > 31 ? 31 : v);
        xoff[s] = ((row * 64 + q) * 32 + uc) * 32 + vc;
        goff[s] = row * 576 + kk;
        loff[s] = row * RS + kk;
        if (ok) vmask |= (1u << s);
    }

    // ---- WMMA fragment addressing (wave32) -------------------------------
    const int lane  = tid & 31;
    const int wv    = tid >> 5;
    const int m0    = (wv & 3) * 16;        // batch tile origin
    const int n0    = (wv >> 2) * 16;       // oc tile origin
    const int lrow  = lane & 15;
    const int khalf = (lane >> 4) * 2;      // lanes 16-31 hold K=2,3

    const float* aLds = &As[(m0 + lrow) * RS + khalf];
    const float* bLds = &Bs[(n0 + lrow) * RS + khalf];

    v8f acc = {};

    // ---- main K loop: stage -> barrier -> 18x wmma -----------------------
    for (int chunk = 0; chunk < NCHUNK; ++chunk) {
        float av[ELEMS], bv[ELEMS];
#pragma unroll
        for (int s = 0; s < ELEMS; ++s) {
            float vx = x[xoff[s] + chunk * 8192];          // +8 channels
            av[s]    = ((vmask >> s) & 1u) ? vx : 0.0f;    // zero padding
            bv[s]    = __builtin_nontemporal_load(wbase + goff[s] + chunk * KC);
        }
        __syncthreads();   // previous chunk's consumers are done
#pragma unroll
        for (int s = 0; s < ELEMS; ++s) {
            As[loff[s]] = av[s];
            Bs[loff[s]] = bv[s];
        }
        __syncthreads();   // staged data visible to all waves
#pragma unroll
        for (int t = 0; t < KC / 4; ++t) {
            v2f a = *(const v2f*)(aLds + t * 4);   // A[m][k], K pair per lane
            v2f b = *(const v2f*)(bLds + t * 4);   // B[k][n] = W[n][k]
            acc = __builtin_amdgcn_wmma_f32_16x16x4_f32(
                false, a, false, b, (short)0, acc, false, false);
        }
    }

    // ---- store D: VGPR r -> (M = r + 8*(lane>=16), N = lane&15) ----------
#pragma unroll
    for (int r = 0; r < 8; ++r) {
        int b  = m0 + r + (lane >> 4) * 8;   // batch
        int oc = n0 + (lane & 15);           // out channel
        out[(size_t)(((b * 64 + oc) << 10) + bid)] = acc[r];
    }
}

extern "C" void kernel_launch(void* const* d_in, const int* in_sizes, int n_in,
                              void* d_out, int out_size, void* d_ws, size_t ws_size,
                              hipStream_t stream) {
    const float* x   = (const float*)d_in[0];   // 64*64*32*32
    const float* wts = (const float*)d_in[1];   // 32*32*64*64*3*3
    float* out       = (float*)d_out;           // 64*64*32*32
    lc2d_wmma_f32<<<dim3(1024), dim3(TB), 0, stream>>>(x, wts, out);
}